// YoloV4Loss_2095944040971
// MI455X (gfx1250) — compile-verified
//
#include <hip/hip_runtime.h>
#include <hip/hip_bf16.h>

typedef __attribute__((ext_vector_type(2))) float v2f;
typedef __attribute__((ext_vector_type(8))) float v8f;
typedef __attribute__((__vector_size__(4 * sizeof(int)))) int v4i;

#define EPS_F   1e-6f
#define K4PI2   0.40528473456935108577f   // 4 / pi^2
#define NBOX    150
#define NANCH   3
#define NCLS    20

#if defined(__has_builtin)
#  if __has_builtin(__builtin_amdgcn_wmma_f32_16x16x4_f32)
#    define HAVE_WMMA_F32X4 1
#  endif
#  if __has_builtin(__builtin_amdgcn_global_load_async_to_lds_b128)
#    define HAVE_ASYNC_LDS 1
#  endif
#  if __has_builtin(__builtin_amdgcn_s_wait_asynccnt)
#    define HAVE_WAIT_ASYNC 1
#  endif
#endif

// ---------------------------------------------------------------------------
// Exact 32-lane sum using V_WMMA_F32_16X16X4_F32 (full fp32 accumulate).
// A layout (ISA 7.12.2, 32-bit A 16x4): VGPR0 lanes0-15 -> A[m=lane][k=0],
// lanes16-31 -> A[m=lane-16][k=2]; VGPR1 -> k=1/k=3 (we feed 0 there).
// B = all-ones  =>  T[m][n] = A[m][0]+A[m][2] = x[m] + x[m+16]  (any n;
// B's layout is irrelevant since it is constant 1).
// D layout: VGPR r holds row r (lanes 0-15) and row r+8 (lanes 16-31), so
// summing t[0..7] gives rows 0..7 on lanes<16 and rows 8..15 on lanes>=16;
// one xor-16 shuffle completes the 32-lane total (identical on all lanes).
// ---------------------------------------------------------------------------
__device__ __forceinline__ float wave_sum32(float x) {
#ifdef HAVE_WMMA_F32X4
  v2f a;  a[0] = x;    a[1] = 0.0f;
  v2f bo; bo[0] = 1.0f; bo[1] = 1.0f;
  v8f c = {};
  v8f t = __builtin_amdgcn_wmma_f32_16x16x4_f32(
      /*neg_a=*/false, a, /*neg_b=*/false, bo,
      /*c_mod=*/(short)0, c, /*reuse_a=*/false, /*reuse_b=*/false);
  float q = ((t[0] + t[1]) + (t[2] + t[3])) + ((t[4] + t[5]) + (t[6] + t[7]));
  q += __shfl_xor(q, 16, 32);
  return q;
#else
  for (int o = 16; o > 0; o >>= 1) x += __shfl_xor(x, o, 32);
  return x;
#endif
}

__device__ __forceinline__ float bce_logits(float x, float t) {
  // logaddexp(0,x) - x*t, numerically stable
  return fmaxf(x, 0.0f) + log1pf(expf(-fabsf(x))) - x * t;
}

// ---------------------------------------------------------------------------
// One kernel per pyramid level. blockIdx.y = batch, 256 threads per block,
// one thread per anchor. The 150 per-batch bboxes are staged (async if the
// builtin exists) and pre-digested once per block into LDS.
// ---------------------------------------------------------------------------
__global__ __launch_bounds__(256) void yolo_layer_kernel(
    const float* __restrict__ p, const float* __restrict__ pd,
    const float* __restrict__ label, const float* __restrict__ bboxes,
    float* __restrict__ partials, int g, float inv_img2)
{
  const int tid = threadIdx.x;
  const int b   = blockIdx.y;
  const int nAnchors = g * g * NANCH;

  __shared__ float4 sBoxA[NBOX];   // lo.x, lo.y, hi.x, hi.y
  __shared__ float4 sBoxB[NBOX];   // cx, cy, area, atan(w/max(h,eps))
  __shared__ float  sWave[8][3];

  float4 raw = make_float4(0.f, 0.f, 0.f, 0.f);
#ifdef HAVE_ASYNC_LDS
  __shared__ float4 sRaw[NBOX];
  if (tid < NBOX) {
    __builtin_amdgcn_global_load_async_to_lds_b128(
        (v4i*)(bboxes + ((size_t)b * NBOX + tid) * 4),
        (v4i*)&sRaw[tid],
        /*offset=*/0, /*cpol=*/0);
  }
#ifdef HAVE_WAIT_ASYNC
  __builtin_amdgcn_s_wait_asynccnt(0);
#else
  asm volatile("s_wait_asynccnt 0" ::: "memory");
#endif
  __syncthreads();
  if (tid < NBOX) raw = sRaw[tid];
#else
  if (tid < NBOX)
    raw = reinterpret_cast<const float4*>(bboxes)[(size_t)b * NBOX + tid];
#endif

  if (tid < NBOX) {
    float cx = raw.x, cy = raw.y, w = raw.z, h = raw.w;
    float ltx = cx - w * 0.5f, rbx = cx + w * 0.5f;
    float lty = cy - h * 0.5f, rby = cy + h * 0.5f;
    float lox = fminf(ltx, rbx), hix = fmaxf(ltx, rbx);
    float loy = fminf(lty, rby), hiy = fmaxf(lty, rby);
    float area = (hix - lox) * (hiy - loy);
    float at = atanf(w / fmaxf(h, EPS_F));
    sBoxA[tid] = make_float4(lox, loy, hix, hiy);
    sBoxB[tid] = make_float4(cx, cy, area, at);
  }
  __syncthreads();

  float c_ciou = 0.0f, c_conf = 0.0f, c_cls = 0.0f;
  const int idx = blockIdx.x * 256 + tid;
  if (idx < nAnchors) {
    const int base = b * nAnchors + idx;
    const float* pdp = pd + (size_t)base * 21;
    const float* pp  = p  + (size_t)base * 25;
    const float* lb  = label + (size_t)base * 26;

    // predicted box, pre-digested
    float x1 = pdp[0], y1 = pdp[1], w1 = pdp[2], h1 = pdp[3];
    float ltx = x1 - w1 * 0.5f, rbx = x1 + w1 * 0.5f;
    float lty = y1 - h1 * 0.5f, rby = y1 + h1 * 0.5f;
    float lo1x = fminf(ltx, rbx), hi1x = fmaxf(ltx, rbx);
    float lo1y = fminf(lty, rby), hi1y = fmaxf(lty, rby);
    float area1 = (hi1x - lo1x) * (hi1y - lo1y);
    float atan1 = atanf(w1 / fmaxf(h1, EPS_F));

    float obj = lb[4], mix = lb[5];

    // ---- CIoU vs label box + class BCE (exact skip when obj == 0) ----
    if (obj != 0.0f) {
      float lx = lb[0], ly = lb[1], lw = lb[2], lh = lb[3];
      float l0x = lx - lw * 0.5f, l1x = lx + lw * 0.5f;
      float l0y = ly - lh * 0.5f, l1y = ly + lh * 0.5f;
      float lo2x = fminf(l0x, l1x), hi2x = fmaxf(l0x, l1x);
      float lo2y = fminf(l0y, l1y), hi2y = fmaxf(l0y, l1y);
      float area2 = (hi2x - lo2x) * (hi2y - lo2y);
      float iw = fmaxf(fminf(hi1x, hi2x) - fmaxf(lo1x, lo2x), 0.0f);
      float ih = fmaxf(fminf(hi1y, hi2y) - fmaxf(lo1y, lo2y), 0.0f);
      float inter = iw * ih;
      float uni = area1 + area2 - inter;
      float iou = inter / fmaxf(uni, EPS_F);
      float ow = fmaxf(hi1x, hi2x) - fminf(lo1x, lo2x);
      float oh = fmaxf(hi1y, hi2y) - fminf(lo1y, lo2y);
      float cc = fmaxf(ow * ow + oh * oh, EPS_F);
      float dx = x1 - lx, dy = y1 - ly;
      float u = (dx * dx + dy * dy) / cc;
      float da = atanf(lw / fmaxf(lh, EPS_F)) - atan1;
      float v = K4PI2 * da * da;
      float al = v / fmaxf(1.0f - iou + v, EPS_F);
      float ciou = iou - (u + al * v);

      float scale = 2.0f - lw * lh * inv_img2;
      c_ciou = obj * scale * (1.0f - ciou) * mix;

      float s = 0.0f;
      #pragma unroll
      for (int c = 0; c < NCLS; ++c) s += bce_logits(pp[5 + c], lb[6 + c]);
      c_cls = obj * s * mix;
    }

    // ---- max CIoU vs 150 bboxes (hot loop; per-bbox terms hoisted to LDS) ----
    float imax = -3.0e38f;
    for (int j = 0; j < NBOX; ++j) {
      float4 A4 = sBoxA[j];      // broadcast LDS reads (no bank conflicts)
      float4 B4 = sBoxB[j];
      float iw = fmaxf(fminf(hi1x, A4.z) - fmaxf(lo1x, A4.x), 0.0f);
      float ih = fmaxf(fminf(hi1y, A4.w) - fmaxf(lo1y, A4.y), 0.0f);
      float inter = iw * ih;
      float uni = area1 + B4.z - inter;
      float iou = __fdividef(inter, fmaxf(uni, EPS_F));
      float ow = fmaxf(hi1x, A4.z) - fminf(lo1x, A4.x);
      float oh = fmaxf(hi1y, A4.w) - fminf(lo1y, A4.y);
      float cc = fmaxf(ow * ow + oh * oh, EPS_F);
      float dx = x1 - B4.x, dy = y1 - B4.y;
      float u = __fdividef(dx * dx + dy * dy, cc);
      float da = B4.w - atan1;
      float v = K4PI2 * da * da;
      float al = __fdividef(v, fmaxf(1.0f - iou + v, EPS_F));
      imax = fmaxf(imax, iou - (u + al * v));
    }

    float noobj = (1.0f - obj) * ((imax < 0.5f) ? 1.0f : 0.0f);
    float xc  = pp[4];
    float sig = 1.0f / (1.0f + expf(-xc));
    float d   = obj - sig;                         // GAMMA=2, ALPHA=1
    float focal = bce_logits(xc, obj) * (d * d);
    c_conf = (obj + noobj) * focal * mix;
  }

  // ---- block reduction: WMMA wave-sum, then fixed-order across 8 waves ----
  float s0 = wave_sum32(c_ciou);
  float s1 = wave_sum32(c_conf);
  float s2 = wave_sum32(c_cls);
  const int lane = tid & 31, wid = tid >> 5;
  if (lane == 0) { sWave[wid][0] = s0; sWave[wid][1] = s1; sWave[wid][2] = s2; }
  __syncthreads();
  if (tid < 3) {
    float s = 0.0f;
    #pragma unroll
    for (int wv = 0; wv < 8; ++wv) s += sWave[wv][tid];
    partials[(size_t)(blockIdx.y * gridDim.x + blockIdx.x) * 3 + tid] = s;
  }
}

// Deterministic fixed-order final reduction of per-block partials.
__global__ __launch_bounds__(128) void yolo_finalize_kernel(
    const float* __restrict__ parts, int nblocks, float* __restrict__ out)
{
  __shared__ float red[3][128];
  const int t = threadIdx.x;
  float s0 = 0.f, s1 = 0.f, s2 = 0.f;
  for (int i = t; i < nblocks; i += 128) {
    s0 += parts[3 * i + 0];
    s1 += parts[3 * i + 1];
    s2 += parts[3 * i + 2];
  }
  red[0][t] = s0; red[1][t] = s1; red[2][t] = s2;
  __syncthreads();
  for (int o = 64; o > 0; o >>= 1) {
    if (t < o) {
      red[0][t] += red[0][t + o];
      red[1][t] += red[1][t + o];
      red[2][t] += red[2][t + o];
    }
    __syncthreads();
  }
  if (t == 0) {
    float a = red[0][0] * 0.25f;   // / bs, bs == 4 for every layer
    float b = red[1][0] * 0.25f;
    float c = red[2][0] * 0.25f;
    out[0] = a + b + c;   // loss
    out[1] = a;           // loss_ciou
    out[2] = b;           // loss_conf
    out[3] = c;           // loss_cls
  }
}

extern "C" void kernel_launch(void* const* d_in, const int* in_sizes, int n_in,
                              void* d_out, int out_size, void* d_ws, size_t ws_size,
                              hipStream_t stream) {
  (void)in_sizes; (void)n_in; (void)out_size; (void)ws_size;
  // setup_inputs() dict order: p0, p_d0, label0, bboxes0, p1, p_d1, label1,
  // bboxes1, p2, p_d2, label2, bboxes2.
  const float* p0  = (const float*)d_in[0];
  const float* pd0 = (const float*)d_in[1];
  const float* lb0 = (const float*)d_in[2];
  const float* bb0 = (const float*)d_in[3];
  const float* p1  = (const float*)d_in[4];
  const float* pd1 = (const float*)d_in[5];
  const float* lb1 = (const float*)d_in[6];
  const float* bb1 = (const float*)d_in[7];
  const float* p2  = (const float*)d_in[8];
  const float* pd2 = (const float*)d_in[9];
  const float* lb2 = (const float*)d_in[10];
  const float* bb2 = (const float*)d_in[11];

  float* parts = (float*)d_ws;
  const float inv_img2 = 1.0f / (608.0f * 608.0f);   // img == 608 for all levels
  const int g0 = 76, g1 = 38, g2 = 19, B = 4;
  const int nb0 = (g0 * g0 * NANCH + 255) / 256;     // 68
  const int nb1 = (g1 * g1 * NANCH + 255) / 256;     // 17
  const int nb2 = (g2 * g2 * NANCH + 255) / 256;     // 5

  yolo_layer_kernel<<<dim3(nb0, B), 256, 0, stream>>>(p0, pd0, lb0, bb0,
      parts, g0, inv_img2);
  yolo_layer_kernel<<<dim3(nb1, B), 256, 0, stream>>>(p1, pd1, lb1, bb1,
      parts + (size_t)nb0 * B * 3, g1, inv_img2);
  yolo_layer_kernel<<<dim3(nb2, B), 256, 0, stream>>>(p2, pd2, lb2, bb2,
      parts + (size_t)(nb0 + nb1) * B * 3, g2, inv_img2);

  const int total_blocks = (nb0 + nb1 + nb2) * B;    // 360
  yolo_finalize_kernel<<<1, 128, 0, stream>>>(parts, total_blocks, (float*)d_out);
}